// ExpertParallelFrontBlock_47863115546643
// MI455X (gfx1250) — compile-verified
//
#include <hip/hip_runtime.h>
#include <hip/hip_bf16.h>

#define S_TOK 4096
#define DDIM  1024
#define NEXP  8
#define FDIM  4096
#define CAP   640   // floor(1.25*4096/8)=640, even, >=4

typedef __attribute__((ext_vector_type(2))) float v2f;
typedef __attribute__((ext_vector_type(8))) float v8f;
typedef __attribute__((ext_vector_type(4))) int   v4i;

// ---------------------------------------------------------------------------
// CDNA5 async global->LDS copy (ASYNCcnt path), guarded for portability.
// Probe-discovered signature: (AS1 v4i* gsrc, AS3 v4i* ldst, imm offset, imm cpol)
// ---------------------------------------------------------------------------
#if defined(__HIP_DEVICE_COMPILE__) && \
    __has_builtin(__builtin_amdgcn_global_load_async_to_lds_b128)
#define USE_ASYNC_LDS 1
__device__ __forceinline__ void async_cp16(const float* gp, void* lp) {
  __builtin_amdgcn_global_load_async_to_lds_b128(
      (__attribute__((address_space(1))) v4i*)(v4i*)(void*)(float*)gp,
      (__attribute__((address_space(3))) v4i*)(v4i*)lp, 0, 0);
}
#else
#define USE_ASYNC_LDS 0
__device__ __forceinline__ void async_cp16(const float* gp, void* lp) {
  *(float4*)lp = *(const float4*)gp;   // synchronous fallback
}
#endif

__device__ __forceinline__ void wait_async_le(int n) {
#if USE_ASYNC_LDS
  if (n == 0)
    asm volatile("s_wait_asynccnt 0x0" ::: "memory");
  else
    asm volatile("s_wait_asynccnt 0x6" ::: "memory");
#endif
}

// ---------------------------------------------------------------------------
// Kernel 1: init router scratch (deterministic every call)
// ---------------------------------------------------------------------------
__global__ void init_router_ws(int* __restrict__ top1_total,
                               int* __restrict__ slot_src) {
  int t = blockIdx.x * blockDim.x + threadIdx.x;
  if (t < NEXP) top1_total[t] = 0;
  if (t < NEXP * CAP) slot_src[t] = -1;
}

// ---------------------------------------------------------------------------
// Kernel 2: gate logits + top-2 expert selection (one wave32 per token).
// softmax skipped: argmax(softmax(z)) == argmax(z), and the returned tensor
// only depends on slot assignment, not on the gate probabilities.
// ---------------------------------------------------------------------------
__global__ __launch_bounds__(256)
void gate_top2_kernel(const float* __restrict__ x,
                      const float* __restrict__ gate_w,
                      int* __restrict__ top1_idx,
                      int* __restrict__ top2_idx,
                      int* __restrict__ top1_total) {
  const int lane = threadIdx.x & 31;
  const int wave = threadIdx.x >> 5;
  const int s = blockIdx.x * 8 + wave;

  float acc[NEXP];
#pragma unroll
  for (int e = 0; e < NEXP; ++e) acc[e] = 0.f;

  const float* xs = x + (size_t)s * DDIM;
  for (int d = lane; d < DDIM; d += 32) {
    float xv = xs[d];
#pragma unroll
    for (int e = 0; e < NEXP; ++e) acc[e] += xv * gate_w[e * DDIM + d];
  }
#pragma unroll
  for (int e = 0; e < NEXP; ++e) {
#pragma unroll
    for (int off = 16; off > 0; off >>= 1)
      acc[e] += __shfl_xor(acc[e], off, 32);
  }
  if (lane == 0) {
    int e1 = 0; float b1v = acc[0];
#pragma unroll
    for (int e = 1; e < NEXP; ++e)
      if (acc[e] > b1v) { b1v = acc[e]; e1 = e; }   // first index on ties
    int e2 = -1; float b2v = -3.4e38f;
#pragma unroll
    for (int e = 0; e < NEXP; ++e)
      if (e != e1 && acc[e] > b2v) { b2v = acc[e]; e2 = e; }
    top1_idx[s] = e1;
    top2_idx[s] = e2;
    atomicAdd(&top1_total[e1], 1);
  }
}

// ---------------------------------------------------------------------------
// Kernel 3: order-preserving per-expert ranking (cumsum semantics) via
// wave32 ballot scan + cross-wave prefix. Single block, 8 waves, 16 chunks.
// rank2 base = total top1 count per expert (reference semantics).
// ---------------------------------------------------------------------------
__global__ __launch_bounds__(256)
void route_rank_kernel(const int* __restrict__ top1_idx,
                       const int* __restrict__ top2_idx,
                       const int* __restrict__ top1_total,
                       int* __restrict__ slot_src) {
  __shared__ int waveCnt1[8][NEXP], waveCnt2[8][NEXP];
  __shared__ int pref1[8][NEXP], pref2[8][NEXP];
  __shared__ int run1[NEXP], run2[NEXP];
  const int tid = threadIdx.x, lane = tid & 31, wave = tid >> 5;
  if (tid < NEXP) { run1[tid] = 0; run2[tid] = top1_total[tid]; }
  __syncthreads();
  const unsigned lt = (1u << lane) - 1u;

  for (int base = 0; base < S_TOK; base += 256) {
    const int s = base + tid;
    const int e1 = top1_idx[s], e2 = top2_idx[s];
    int off1 = 0, off2 = 0;
#pragma unroll
    for (int e = 0; e < NEXP; ++e) {
      unsigned m1 = (unsigned)__ballot(e1 == e);
      unsigned m2 = (unsigned)__ballot(e2 == e);
      if (e == e1) off1 = __popc(m1 & lt);
      if (e == e2) off2 = __popc(m2 & lt);
      if (lane == e) { waveCnt1[wave][e] = __popc(m1); waveCnt2[wave][e] = __popc(m2); }
    }
    __syncthreads();
    if (tid < NEXP) {
      const int e = tid;
      int r1 = run1[e], r2 = run2[e];
#pragma unroll
      for (int w = 0; w < 8; ++w) {
        pref1[w][e] = r1; r1 += waveCnt1[w][e];
        pref2[w][e] = r2; r2 += waveCnt2[w][e];
      }
      run1[e] = r1; run2[e] = r2;
    }
    __syncthreads();
    const int rank1 = pref1[wave][e1] + off1;
    if (rank1 < CAP) slot_src[e1 * CAP + rank1] = s;
    const int rank2 = pref2[wave][e2] + off2;
    if (rank2 < CAP) slot_src[e2 * CAP + rank2] = s;
    __syncthreads();
  }
}

// ---------------------------------------------------------------------------
// Kernel 4: fused gather + batched GEMM + bias via V_WMMA_F32_16X16X4_F32,
// double-buffered LDS fed by GLOBAL_LOAD_ASYNC_TO_LDS_B128 (ASYNCcnt).
// out[e, c, f] = sum_d x[slot_src[e][c], d] * weight[e, d, f] + bias[e, f]
// Empty slots: gather a dummy row (uniform async count) and emit bias only.
// BM=64 BN=128 BK=32, 256 threads = 8 waves (2x4), wave tile 32x32 (2x2 WMMA)
// ---------------------------------------------------------------------------
#define BM 64
#define BN 128
#define BK 32
#define LDA 36    // padded row stride (floats), 16B-aligned rows
#define LDB 132   // padded row stride (floats), 16B-aligned rows
#define NKB (DDIM / BK)   // 32 K-blocks

__global__ __launch_bounds__(256)
void moe_ffn_wmma_kernel(const float* __restrict__ x,
                         const float* __restrict__ weight,
                         const float* __restrict__ bias,
                         const int* __restrict__ slot_src,
                         float* __restrict__ out) {
  __shared__ float ALDS[2][BM][LDA];
  __shared__ float BLDS[2][BK][LDB];
  __shared__ int srcRow[BM];

  const int e = blockIdx.z;
  const int mBase = blockIdx.y * BM;
  const int nBase = blockIdx.x * BN;
  const int tid = threadIdx.x;
  const int lane = tid & 31, wave = tid >> 5;
  const int wy = wave >> 2, wx = wave & 3;   // 2 x 4 wave grid
  const int m0 = wy * 32, n0 = wx * 32;
  const int hi = lane >> 4;                  // lane group 0/1
  const int lx = lane & 15;

  if (tid < BM) srcRow[tid] = slot_src[e * CAP + mBase + tid];
  __syncthreads();

  // per-thread staging coordinates (fixed across K-blocks)
  const int aRow0 = tid >> 3,         aQ0 = (tid & 7) * 4;           // idx = tid
  const int aRow1 = (tid + 256) >> 3, aQ1 = ((tid + 256) & 7) * 4;   // idx = tid+256
  const int aSrc0 = max(srcRow[aRow0], 0);   // clamp: garbage rows discarded later
  const int aSrc1 = max(srcRow[aRow1], 0);
  const float* wbase = weight + (size_t)e * DDIM * FDIM + nBase;

  v8f acc[2][2];
#pragma unroll
  for (int mi = 0; mi < 2; ++mi)
#pragma unroll
    for (int ni = 0; ni < 2; ++ni)
#pragma unroll
      for (int k = 0; k < 8; ++k) acc[mi][ni][k] = 0.f;

  // ---- staging: exactly 6 async b128 per wave per stage (2 A + 4 B) ----
  auto stage = [&](int buf, int k0) {
    async_cp16(x + (size_t)aSrc0 * DDIM + k0 + aQ0, &ALDS[buf][aRow0][aQ0]);
    async_cp16(x + (size_t)aSrc1 * DDIM + k0 + aQ1, &ALDS[buf][aRow1][aQ1]);
#pragma unroll
    for (int i = 0; i < 4; ++i) {
      const int idx = tid + 256 * i;
      const int row = idx >> 5, q = (idx & 31) * 4;
      async_cp16(wbase + (size_t)(k0 + row) * FDIM + q, &BLDS[buf][row][q]);
    }
  };

  stage(0, 0);   // prologue

  for (int it = 0; it < NKB; ++it) {
    const int cur = it & 1;
    const bool more = (it + 1) < NKB;
    if (more) stage(cur ^ 1, (it + 1) * BK);   // overlap next tile with compute
    wait_async_le(more ? 6 : 0);   // in-order completion: prev stage landed
    __syncthreads();

    const float (*A)[LDA] = ALDS[cur];
    const float (*B)[LDB] = BLDS[cur];
#pragma unroll
    for (int kk = 0; kk < BK; kk += 4) {
      // A frag 16x4: lane L holds A[M=L%16][2*(L/16) + {0,1}] (contiguous pair)
      v2f afrag[2];
#pragma unroll
      for (int mi = 0; mi < 2; ++mi)
        afrag[mi] = *(const v2f*)&A[m0 + mi * 16 + lx][kk + hi * 2];
      // B frag 4x16: lane L holds B[2*(L/16) + {0,1}][N=L%16]
      v2f bfrag[2];
#pragma unroll
      for (int ni = 0; ni < 2; ++ni) {
        v2f b;
        b[0] = B[kk + hi * 2 + 0][n0 + ni * 16 + lx];
        b[1] = B[kk + hi * 2 + 1][n0 + ni * 16 + lx];
        bfrag[ni] = b;
      }
#pragma unroll
      for (int mi = 0; mi < 2; ++mi)
#pragma unroll
        for (int ni = 0; ni < 2; ++ni)
          acc[mi][ni] = __builtin_amdgcn_wmma_f32_16x16x4_f32(
              false, afrag[mi], false, bfrag[ni],
              (short)0, acc[mi][ni], false, false);
    }
    __syncthreads();   // all waves done reading buf before it is re-staged
  }

  // ---- epilogue: C layout VGPR i -> M = i + 8*hi, N = lx; fuse bias.
  //      Empty slots (src<0) accumulated a dummy row: emit bias only. ----
#pragma unroll
  for (int ni = 0; ni < 2; ++ni) {
    const int col = nBase + n0 + ni * 16 + lx;
    const float bv = bias[e * FDIM + col];
#pragma unroll
    for (int mi = 0; mi < 2; ++mi) {
#pragma unroll
      for (int i = 0; i < 8; ++i) {
        const int rl = m0 + mi * 16 + i + hi * 8;
        const float v = (srcRow[rl] >= 0) ? (acc[mi][ni][i] + bv) : bv;
        out[((size_t)e * CAP + mBase + rl) * FDIM + col] = v;
      }
    }
  }
}

// ---------------------------------------------------------------------------
extern "C" void kernel_launch(void* const* d_in, const int* in_sizes, int n_in,
                              void* d_out, int out_size, void* d_ws, size_t ws_size,
                              hipStream_t stream) {
  const float* x      = (const float*)d_in[0];   // [S, D]
  const float* gate_w = (const float*)d_in[1];   // [E, D]
  const float* weight = (const float*)d_in[2];   // [E, D, F]
  const float* bias   = (const float*)d_in[3];   // [E, 1, F]
  float* out = (float*)d_out;                    // [E, CAP, F] fp32

  int* top1_idx   = (int*)d_ws;                  // S
  int* top2_idx   = top1_idx + S_TOK;            // S
  int* top1_total = top2_idx + S_TOK;            // NEXP
  int* slot_src   = top1_total + NEXP;           // NEXP*CAP

  init_router_ws<<<(NEXP * CAP + 255) / 256, 256, 0, stream>>>(top1_total, slot_src);
  gate_top2_kernel<<<S_TOK / 8, 256, 0, stream>>>(x, gate_w, top1_idx, top2_idx, top1_total);
  route_rank_kernel<<<1, 256, 0, stream>>>(top1_idx, top2_idx, top1_total, slot_src);

  dim3 grid(FDIM / BN, CAP / BM, NEXP);          // 32 x 10 x 8 = 2560 blocks
  moe_ffn_wmma_kernel<<<grid, 256, 0, stream>>>(x, weight, bias, slot_src, out);
}